// CE_OHEM_14998025797700
// MI455X (gfx1250) — compile-verified
//
#include <hip/hip_runtime.h>

typedef __attribute__((ext_vector_type(2))) float v2f;
typedef __attribute__((ext_vector_type(8))) float v8f;
typedef __attribute__((ext_vector_type(4))) float f32x4;
typedef __attribute__((ext_vector_type(2))) float f32x2;
typedef __attribute__((ext_vector_type(4))) int   i32x4;

#define N_PIX   (8 * 512 * 1024)   // 4194304 pixels
#define NCLS    20
#define PLANE   (512 * 1024)       // 524288, HW plane size
#define KTOP    1258291            // int(N_PIX * 0.3)
#define NB_FIN  1024               // blocks in final pass (N_PIX / NB_FIN = 4096)

struct SelState { unsigned prefix; unsigned krem; };

// ---- monotonic f32 <-> u32 key mapping (ascending order preserved) ----
__device__ __forceinline__ unsigned fkey(float f) {
    unsigned b = __float_as_uint(f);
    return (b & 0x80000000u) ? ~b : (b | 0x80000000u);
}
__device__ __forceinline__ float keyToFloat(unsigned k) {
    unsigned b = (k & 0x80000000u) ? (k ^ 0x80000000u) : ~k;
    return __uint_as_float(b);
}

// ---------------- init: zero hist / sel / count ----------------
__global__ void ohem_init_kernel(unsigned* hist, SelState* sel, unsigned* cnt) {
    int t = threadIdx.x;
    if (t < 256) hist[t] = 0u;
    if (t == 0) { sel->prefix = 0u; sel->krem = (unsigned)KTOP; *cnt = 0u; }
}

// ---------------- pass 1: per-pixel cross entropy ----------------
// pred [B, C, H, W] f32 (336 MB, streamed once -> non-temporal so the 16 MB
// CE array stays resident in the 192 MB L2 for the later select passes).
__global__ void ohem_ce_kernel(const float* __restrict__ pred,
                               const int* __restrict__ target,
                               float* __restrict__ ce) {
    int idx = blockIdx.x * blockDim.x + threadIdx.x;   // 0 .. N_PIX/4 - 1
    int i = idx * 4;
    int b  = i >> 19;            // / PLANE
    int hw = i & (PLANE - 1);
    const float* base = pred + (size_t)b * NCLS * PLANE + hw;

    f32x4 x[NCLS];
#pragma unroll
    for (int c = 0; c < NCLS; ++c)
        x[c] = __builtin_nontemporal_load((const f32x4*)(base + (size_t)c * PLANE));

    f32x4 mx = x[0];
#pragma unroll
    for (int c = 1; c < NCLS; ++c) {
        mx.x = fmaxf(mx.x, x[c].x); mx.y = fmaxf(mx.y, x[c].y);
        mx.z = fmaxf(mx.z, x[c].z); mx.w = fmaxf(mx.w, x[c].w);
    }

    i32x4 tv = __builtin_nontemporal_load((const i32x4*)(target + i));
    f32x4 xt = x[0];
#pragma unroll
    for (int c = 1; c < NCLS; ++c) {
        xt.x = (tv.x == c) ? x[c].x : xt.x;
        xt.y = (tv.y == c) ? x[c].y : xt.y;
        xt.z = (tv.z == c) ? x[c].z : xt.z;
        xt.w = (tv.w == c) ? x[c].w : xt.w;
    }

    f32x4 s = {0.f, 0.f, 0.f, 0.f};
#pragma unroll
    for (int c = 0; c < NCLS; ++c) {
        s.x += __expf(x[c].x - mx.x); s.y += __expf(x[c].y - mx.y);
        s.z += __expf(x[c].z - mx.z); s.w += __expf(x[c].w - mx.w);
    }

    f32x4 r;
    r.x = __logf(s.x) + mx.x - xt.x;
    r.y = __logf(s.y) + mx.y - xt.y;
    r.z = __logf(s.z) + mx.z - xt.z;
    r.w = __logf(s.w) + mx.w - xt.w;
    *(f32x4*)(ce + i) = r;      // regular (RT) store: keep CE in L2
}

// ---------------- radix-select histogram pass (one byte) ----------------
// CE is L2-resident; read it as B128 to minimize instruction count.
__global__ void ohem_hist_kernel(const float* __restrict__ ce,
                                 unsigned* __restrict__ hist,
                                 const SelState* __restrict__ sel,
                                 int shift) {
    __shared__ unsigned lh[256];
    int tid = threadIdx.x;
    lh[tid] = 0u;
    __syncthreads();

    unsigned highmask = (shift == 24) ? 0u : (0xFFFFFFFFu << (shift + 8));
    unsigned prefix = sel->prefix;
    int stride = gridDim.x * blockDim.x;               // 65536 threads
    for (int q = blockIdx.x * blockDim.x + tid; q < N_PIX / 4; q += stride) {
        f32x4 v = *(const f32x4*)(ce + q * 4);
        unsigned k0 = fkey(v.x), k1 = fkey(v.y), k2 = fkey(v.z), k3 = fkey(v.w);
        if (((k0 ^ prefix) & highmask) == 0u) atomicAdd(&lh[(k0 >> shift) & 255u], 1u);
        if (((k1 ^ prefix) & highmask) == 0u) atomicAdd(&lh[(k1 >> shift) & 255u], 1u);
        if (((k2 ^ prefix) & highmask) == 0u) atomicAdd(&lh[(k2 >> shift) & 255u], 1u);
        if (((k3 ^ prefix) & highmask) == 0u) atomicAdd(&lh[(k3 >> shift) & 255u], 1u);
    }
    __syncthreads();
    atomicAdd(&hist[tid], lh[tid]);
}

// ---------------- pick the byte holding the k-th largest; reset hist ----------------
__global__ void ohem_select_kernel(unsigned* hist, SelState* sel, int shift) {
    if (threadIdx.x != 0 || blockIdx.x != 0) return;
    unsigned krem = sel->krem;
    unsigned prefix = sel->prefix;
    for (int b = 255; b >= 0; --b) {
        unsigned c = hist[b];
        if (krem <= c) { prefix |= ((unsigned)b) << shift; break; }
        krem -= c;
    }
    sel->prefix = prefix;
    sel->krem = krem;
    for (int b = 0; b < 256; ++b) hist[b] = 0u;   // clean for next round
}

// ---------------- final pass: WMMA ones-matmul reductions ----------------
// Each block owns a contiguous 4096-element chunk; EXEC stays all-ones
// (selects only, exact divisibility) as required for V_WMMA.
// Last reader of CE -> non-temporal loads (discard from cache after use).
__global__ void ohem_final_pass_kernel(const float* __restrict__ ce,
                                       const SelState* __restrict__ sel,
                                       float* __restrict__ partA,
                                       float* __restrict__ partG,
                                       unsigned* __restrict__ cnt) {
    const int tid = threadIdx.x;
    const unsigned tkey = sel->prefix;

    v2f ones; ones[0] = 1.0f; ones[1] = 1.0f;
    v8f accA = {0.f, 0.f, 0.f, 0.f, 0.f, 0.f, 0.f, 0.f};
    v8f accG = {0.f, 0.f, 0.f, 0.f, 0.f, 0.f, 0.f, 0.f};
    unsigned myCnt = 0;

    const int blockBase = blockIdx.x * 4096;
#pragma unroll
    for (int it = 0; it < 8; ++it) {
        int i = blockBase + it * 512 + tid * 2;
        f32x2 p = __builtin_nontemporal_load((const f32x2*)(ce + i));
        float a0 = p.x, a1 = p.y;
        unsigned k0 = fkey(a0), k1 = fkey(a1);
        float g0 = (k0 > tkey) ? a0 : 0.0f;
        float g1 = (k1 > tkey) ? a1 : 0.0f;
        myCnt += (k0 > tkey) ? 1u : 0u;
        myCnt += (k1 > tkey) ? 1u : 0u;
        v2f bA; bA[0] = a0; bA[1] = a1;
        v2f bG; bG[0] = g0; bG[1] = g1;
        // D = ones(16x4) * B(4x16) + D : every input counted exactly 16x in sum(D)
        accA = __builtin_amdgcn_wmma_f32_16x16x4_f32(false, ones, false, bA,
                                                     (short)0, accA, false, false);
        accG = __builtin_amdgcn_wmma_f32_16x16x4_f32(false, ones, false, bG,
                                                     (short)0, accG, false, false);
    }

    float pa = 0.f, pg = 0.f;
#pragma unroll
    for (int j = 0; j < 8; ++j) { pa += accA[j]; pg += accG[j]; }

    __shared__ float sA[256];
    __shared__ float sG[256];
    __shared__ unsigned sC[256];
    sA[tid] = pa; sG[tid] = pg; sC[tid] = myCnt;
    __syncthreads();
    for (int off = 128; off > 0; off >>= 1) {
        if (tid < off) {
            sA[tid] += sA[tid + off];
            sG[tid] += sG[tid + off];
            sC[tid] += sC[tid + off];
        }
        __syncthreads();
    }
    if (tid == 0) {
        partA[blockIdx.x] = sA[0] * (1.0f / 16.0f);   // undo ones-matmul 16x fanout
        partG[blockIdx.x] = sG[0] * (1.0f / 16.0f);
        atomicAdd(cnt, sC[0]);                        // integer: order-independent
    }
}

// ---------------- finalize: fixed-order partial reduce + scalar ----------------
__global__ void ohem_finalize_kernel(const float* __restrict__ partA,
                                     const float* __restrict__ partG,
                                     const unsigned* __restrict__ cnt,
                                     const SelState* __restrict__ sel,
                                     float* __restrict__ out) {
    __shared__ float sA[256];
    __shared__ float sG[256];
    int tid = threadIdx.x;
    float a = 0.f, g = 0.f;
    for (int i = tid; i < NB_FIN; i += 256) { a += partA[i]; g += partG[i]; }
    sA[tid] = a; sG[tid] = g;
    __syncthreads();
    for (int off = 128; off > 0; off >>= 1) {
        if (tid < off) { sA[tid] += sA[tid + off]; sG[tid] += sG[tid + off]; }
        __syncthreads();
    }
    if (tid == 0) {
        float T = keyToFloat(sel->prefix);            // exact k-th largest value
        unsigned cg = *cnt;                           // strictly-greater count
        float topk_sum = sG[0] + (float)((long long)KTOP - (long long)cg) * T;
        float loss = sA[0] / ((float)N_PIX + 1e-12f) + topk_sum / (float)KTOP;
        out[0] = loss;
    }
}

extern "C" void kernel_launch(void* const* d_in, const int* in_sizes, int n_in,
                              void* d_out, int out_size, void* d_ws, size_t ws_size,
                              hipStream_t stream) {
    (void)in_sizes; (void)n_in; (void)out_size; (void)ws_size;

    const float* pred  = (const float*)d_in[0];
    const int*   tgt   = (const int*)d_in[1];
    float*       out   = (float*)d_out;

    char* w = (char*)d_ws;
    const size_t CE_BYTES = (size_t)N_PIX * sizeof(float);   // 16 MB, L2-resident
    float*    ce   = (float*)w;
    unsigned* hist = (unsigned*)(w + CE_BYTES);              // 256 u32
    SelState* sel  = (SelState*)(w + CE_BYTES + 1024);
    unsigned* cnt  = (unsigned*)(w + CE_BYTES + 1040);
    float*    pA   = (float*)(w + CE_BYTES + 2048);          // NB_FIN floats
    float*    pG   = (float*)(w + CE_BYTES + 2048 + 4096);   // NB_FIN floats

    ohem_init_kernel<<<1, 256, 0, stream>>>(hist, sel, cnt);
    ohem_ce_kernel<<<N_PIX / 4 / 256, 256, 0, stream>>>(pred, tgt, ce);

    const int shifts[4] = {24, 16, 8, 0};
    for (int p = 0; p < 4; ++p) {
        ohem_hist_kernel<<<256, 256, 0, stream>>>(ce, hist, sel, shifts[p]);
        ohem_select_kernel<<<1, 1, 0, stream>>>(hist, sel, shifts[p]);
    }

    ohem_final_pass_kernel<<<NB_FIN, 256, 0, stream>>>(ce, sel, pA, pG, cnt);
    ohem_finalize_kernel<<<1, 256, 0, stream>>>(pA, pG, cnt, sel, out);
}